// Router_42597485642410
// MI455X (gfx1250) — compile-verified
//
#include <hip/hip_runtime.h>
#include <hip/hip_bf16.h>
#include <math.h>

typedef float v2f __attribute__((ext_vector_type(2)));
typedef float v8f __attribute__((ext_vector_type(8)));

#define IN_DIM 1024
#define HIDDEN 256
#define KOUT   64
#define MROWS  32
#define KC     16
#define NTOPK  3

// LDS layout (union, phases separated by barriers):
//  phase1 (x2 ping-pong): A tile (32x16 feat chunk, stride 28 -> 16B-aligned rows,
//          conflict-free frag reads), B tile (16x256 W1 chunk, stride 260)
//  phase2: h (32x256 gelu output, stride 260), w2c (64x64 W2 chunk, stride 68)
//  phase3: lg (32x68 logits/probs scratch) overlays everything
struct P1 { float A[MROWS][28]; float B[KC][260]; };
struct P2 { float h[MROWS][260]; float w2c[64][68]; };
union SharedU { P1 p1[2]; P2 p2; float lg[MROWS * 68]; };

// CDNA5 async copy: global -> LDS, tracked by ASYNCcnt (no VGPR round trip).
// GVS addressing: SGPR 64-bit base + per-lane 32-bit byte offset.
__device__ __forceinline__ void async_b128(unsigned lds_addr, unsigned goff_bytes,
                                           const void* sbase) {
  asm volatile("global_load_async_to_lds_b128 %0, %1, %2"
               :: "v"(lds_addr), "v"(goff_bytes), "s"(sbase)
               : "memory");
}
__device__ __forceinline__ void wait_async0() {
  asm volatile("s_wait_asynccnt 0x0" ::: "memory");
}
__device__ __forceinline__ unsigned lds_off(const void* p) {
  return (unsigned)(uintptr_t)p;   // generic LDS addr: low 32 bits = group offset
}

__global__ __launch_bounds__(256) void router_fused(
    const float* __restrict__ feat, const float* __restrict__ W1,
    const float* __restrict__ b1,   const float* __restrict__ W2,
    const float* __restrict__ b2,   float* __restrict__ out, int N)
{
  __shared__ __align__(16) SharedU sh;
  __shared__ float b1s[HIDDEN];
  __shared__ float b2s[KOUT];

  const int tid  = threadIdx.x;
  const int wave = tid >> 5;
  const int lane = tid & 31;
  const int l16  = lane & 15;
  const int koff = (lane < 16) ? 0 : 2;   // ISA 16x16x4 f32 A/B lane->K mapping
  const long long rowBase = (long long)blockIdx.x * MROWS;

  b1s[tid] = b1[tid];
  if (tid < KOUT) b2s[tid] = b2[tid];

  // stage one K-chunk (feat 32x16, W1 16x256) into buffer `buf` via async DMA
  auto stage = [&](int buf, int kc) {
    if (tid < 128) {                         // 128 b128 slots for the feat tile
      int r = tid >> 2, c4 = tid & 3;
      async_b128(lds_off(&sh.p1[buf].A[r][c4 * 4]),
                 (unsigned)(((rowBase + r) * IN_DIM + kc + c4 * 4) * 4), feat);
    }
#pragma unroll
    for (int i = 0; i < 4; ++i) {            // 1024 b128 slots for the W1 chunk
      int s = tid + i * 256;
      int r = s >> 6, c4 = s & 63;
      async_b128(lds_off(&sh.p1[buf].B[r][c4 * 4]),
                 (unsigned)(((kc + r) * HIDDEN + c4 * 4) * 4), W1);
    }
  };

  // ---------------- Phase 1: h = gelu(feat @ W1 + b1) ----------------
  const int rs  = wave >> 2;                 // row sub-tile 0..1
  const int ct0 = (wave & 3) * 4;            // 4 col tiles (64 of 256 cols) per wave
  const int arow = rs * 16 + l16;

  v8f acc[4] = {};

  stage(0, 0);
  const int NCHUNK = IN_DIM / KC;            // 64
  for (int c = 0; c < NCHUNK; ++c) {
    wait_async0();                           // chunk c resident
    __syncthreads();                         // block-wide; prior buffer reads retired
    if (c + 1 < NCHUNK) stage((c + 1) & 1, (c + 1) * KC);  // prefetch next chunk
    const P1& B = sh.p1[c & 1];
#pragma unroll
    for (int ks = 0; ks < KC; ks += 4) {
      v2f a;
      a[0] = B.A[arow][ks + koff];
      a[1] = B.A[arow][ks + koff + 1];
#pragma unroll
      for (int t = 0; t < 4; ++t) {
        int n = (ct0 + t) * 16 + l16;
        v2f bb;
        bb[0] = B.B[ks + koff][n];
        bb[1] = B.B[ks + koff + 1][n];
        acc[t] = __builtin_amdgcn_wmma_f32_16x16x4_f32(
            false, a, false, bb, (short)0, acc[t], false, false);
      }
    }
  }

  __syncthreads();   // all phase-1 LDS reads done before overlaying with h
#pragma unroll
  for (int t = 0; t < 4; ++t) {
    int n = (ct0 + t) * 16 + l16;
    float bias = b1s[n];
#pragma unroll
    for (int i = 0; i < 8; ++i) {
      int r = rs * 16 + ((lane < 16) ? i : i + 8);   // C/D VGPR->row mapping
      float x = acc[t][i] + bias;
      // exact GELU: 0.5*x*(1+erf(x/sqrt(2)))
      sh.p2.h[r][n] = 0.5f * x * (1.0f + erff(x * 0.70710678118654752440f));
    }
  }
  __syncthreads();

  // ---------------- Phase 2: logits = h @ W2 + b2 ----------------
  const int rs2   = wave >> 2;
  const int ct2   = wave & 3;
  const int arow2 = rs2 * 16 + l16;
  const int n2    = ct2 * 16 + l16;
  v8f acc2 = {};

  for (int kk = 0; kk < HIDDEN; kk += 64) {
    __syncthreads();                         // previous w2c chunk fully consumed
#pragma unroll
    for (int i = 0; i < 4; ++i) {            // W2 chunk 64x64, async b128
      int s = tid + i * 256;
      int r = s >> 4, c4 = s & 15;
      async_b128(lds_off(&sh.p2.w2c[r][c4 * 4]),
                 (unsigned)(((kk + r) * KOUT + c4 * 4) * 4), W2);
    }
    wait_async0();
    __syncthreads();
#pragma unroll
    for (int ks = 0; ks < 64; ks += 4) {
      v2f a, bb;
      a[0]  = sh.p2.h[arow2][kk + ks + koff];
      a[1]  = sh.p2.h[arow2][kk + ks + koff + 1];
      bb[0] = sh.p2.w2c[ks + koff][n2];
      bb[1] = sh.p2.w2c[ks + koff + 1][n2];
      acc2 = __builtin_amdgcn_wmma_f32_16x16x4_f32(
          false, a, false, bb, (short)0, acc2, false, false);
    }
  }

  __syncthreads();   // done reading h/w2c; safe to overlay with logits scratch
  {
    float bias = b2s[n2];
#pragma unroll
    for (int i = 0; i < 8; ++i) {
      int r = rs2 * 16 + ((lane < 16) ? i : i + 8);
      sh.lg[r * 68 + n2] = acc2[i] + bias;
    }
  }
  __syncthreads();

  // ------- Phase 3: softmax -> soft top-k -> softmax, 8 lanes per row -------
  // 256 threads = 32 rows x 8 lanes; each lane owns 8 of the 64 columns.
  // Reductions via __shfl_xor over the 8-lane group (stays inside a wave32).
  {
    const int r   = tid >> 3;                // row 0..31
    const int sub = tid & 7;                 // lane-in-row 0..7
    const long long grow = rowBase + r;
    const float* lrow = &sh.lg[r * 68];
    float* out_pused = out;
    float* out_idx   = out + (long long)N * KOUT;
    float* out_psoft = out + (long long)N * KOUT + (long long)N * NTOPK;

    float v[8];
#pragma unroll
    for (int q = 0; q < 8; ++q) v[q] = lrow[sub * 8 + q];

    // p_soft = softmax(logits)
    float m = v[0];
#pragma unroll
    for (int q = 1; q < 8; ++q) m = fmaxf(m, v[q]);
#pragma unroll
    for (int d = 1; d < 8; d <<= 1) m = fmaxf(m, __shfl_xor(m, d, 32));
    float s = 0.0f;
#pragma unroll
    for (int q = 0; q < 8; ++q) { v[q] = expf(v[q] - m); s += v[q]; }
#pragma unroll
    for (int d = 1; d < 8; d <<= 1) s += __shfl_xor(s, d, 32);
    float inv = 1.0f / s;
#pragma unroll
    for (int q = 0; q < 8; ++q) {
      v[q] *= inv;
      out_psoft[grow * KOUT + sub * 8 + q] = v[q];
    }

    // top-3 of p^4 == top-3 of p (x^4 monotone on clamped p).
    // Argmax reduction carrying (value, index); ties -> lowest index (lax.top_k).
    int c0, c1, c2;
    {
      float bv = -1.0f; int bi = 1 << 30;
#pragma unroll
      for (int q = 0; q < 8; ++q) {
        int j = sub * 8 + q;
        if (v[q] > bv) { bv = v[q]; bi = j; }
      }
#pragma unroll
      for (int d = 1; d < 8; d <<= 1) {
        float ov = __shfl_xor(bv, d, 32);
        int   oi = __shfl_xor(bi, d, 32);
        if (ov > bv || (ov == bv && oi < bi)) { bv = ov; bi = oi; }
      }
      c0 = bi;
    }
    {
      float bv = -1.0f; int bi = 1 << 30;
#pragma unroll
      for (int q = 0; q < 8; ++q) {
        int j = sub * 8 + q;
        if (j != c0 && v[q] > bv) { bv = v[q]; bi = j; }
      }
#pragma unroll
      for (int d = 1; d < 8; d <<= 1) {
        float ov = __shfl_xor(bv, d, 32);
        int   oi = __shfl_xor(bi, d, 32);
        if (ov > bv || (ov == bv && oi < bi)) { bv = ov; bi = oi; }
      }
      c1 = bi;
    }
    {
      float bv = -1.0f; int bi = 1 << 30;
#pragma unroll
      for (int q = 0; q < 8; ++q) {
        int j = sub * 8 + q;
        if (j != c0 && j != c1 && v[q] > bv) { bv = v[q]; bi = j; }
      }
#pragma unroll
      for (int d = 1; d < 8; d <<= 1) {
        float ov = __shfl_xor(bv, d, 32);
        int   oi = __shfl_xor(bi, d, 32);
        if (ov > bv || (ov == bv && oi < bi)) { bv = ov; bi = oi; }
      }
      c2 = bi;
    }
    if (sub == 0) {
      out_idx[grow * 3 + 0] = (float)c0;
      out_idx[grow * 3 + 1] = (float)c1;
      out_idx[grow * 3 + 2] = (float)c2;
    }

    // logits2 = (log(max(p,eps)^4 + eps) + (1-mask)*log(beta)) / T ; softmax
    const float logBeta = -4.60517018598809136804f;   // log(0.01)
    const float invT = 10.0f;                         // 1/0.1
    float l2v[8];
    float m2 = -INFINITY;
#pragma unroll
    for (int q = 0; q < 8; ++q) {
      int j = sub * 8 + q;
      float p  = fmaxf(v[q], 1e-12f);
      float p2 = p * p;
      float l2 = logf(p2 * p2 + 1e-12f);
      if (j != c0 && j != c1 && j != c2) l2 += logBeta;
      l2 *= invT;
      l2v[q] = l2;
      m2 = fmaxf(m2, l2);
    }
#pragma unroll
    for (int d = 1; d < 8; d <<= 1) m2 = fmaxf(m2, __shfl_xor(m2, d, 32));
    float s2 = 0.0f;
#pragma unroll
    for (int q = 0; q < 8; ++q) { l2v[q] = expf(l2v[q] - m2); s2 += l2v[q]; }
#pragma unroll
    for (int d = 1; d < 8; d <<= 1) s2 += __shfl_xor(s2, d, 32);
    float inv2 = 1.0f / s2;
#pragma unroll
    for (int q = 0; q < 8; ++q)
      out_pused[grow * KOUT + sub * 8 + q] = l2v[q] * inv2;
  }
}

extern "C" void kernel_launch(void* const* d_in, const int* in_sizes, int n_in,
                              void* d_out, int out_size, void* d_ws, size_t ws_size,
                              hipStream_t stream) {
  (void)n_in; (void)out_size; (void)d_ws; (void)ws_size;
  const float* feat = (const float*)d_in[0];
  const float* W1   = (const float*)d_in[1];
  const float* b1   = (const float*)d_in[2];
  const float* W2   = (const float*)d_in[3];
  const float* b2   = (const float*)d_in[4];
  float* out = (float*)d_out;

  const int N = in_sizes[0] / IN_DIM;       // 131072
  dim3 grid(N / MROWS);                     // 4096 blocks
  router_fused<<<grid, 256, 0, stream>>>(feat, W1, b1, W2, b2, out, N);
}